// Top_ECELoss_51737176047892
// MI455X (gfx1250) — compile-verified
//
#include <hip/hip_runtime.h>
#include <hip/hip_bf16.h>

// ---------------------------------------------------------------------------
// ECE loss for MI455X (gfx1250).
// Memory-bound: 400MB logits @ 23.3 TB/s => ~17us floor. Strategy:
//  - Double-buffered tiles (32 rows x C floats) staged into LDS with the
//    CDNA5 async DMA path: global_load_async_to_lds_b128 + s_wait_asynccnt.
//  - One wave32 per row: butterfly shfl reductions for max/argmax/sum-exp.
//  - Block-local LDS histogram (ds_add_f32 atomics), flushed with global
//    fp32 atomics into d_ws, tiny finalize kernel computes sum|dc-da|/N.
// ---------------------------------------------------------------------------

#define BLOCK      256
#define WAVES      (BLOCK / 32)
#define TILE_ROWS  32
#define MAX_CELLS  2048            // supports num_classes*n_bins up to 2048 (100*15=1500)

__device__ __forceinline__ void async_copy_b128(unsigned lds_addr, const void* gaddr) {
    // CDNA5 async Global->LDS DMA (VGLOBAL encoding, GV mode, tracked by ASYNCcnt).
    // Per lane: copies 16 bytes from the global address in gaddr to the LDS address in lds_addr
    asm volatile("global_load_async_to_lds_b128 %0, %1, off"
                 :: "v"(lds_addr), "v"(gaddr)
                 : "memory");
}

__device__ __forceinline__ void wait_async0() {
    asm volatile("s_wait_asynccnt 0" ::: "memory");
}

extern "C" __global__ void __launch_bounds__(BLOCK)
ece_main(const float* __restrict__ logits, const int* __restrict__ labels,
         const int* __restrict__ nbins_p, int C, int Nrows,
         float* __restrict__ gacc)
{
    extern __shared__ float tiles[];            // 2 * TILE_ROWS * C floats (double buffer)
    __shared__ float hist[2 * MAX_CELLS];       // [0..MAX_CELLS) conf, [MAX_CELLS..) acc

    const int tid  = threadIdx.x;
    const int lane = tid & 31;
    const int wid  = tid >> 5;

    const int nb = *nbins_p;
    int cells = C * nb;
    if (cells > MAX_CELLS) cells = MAX_CELLS;

    const int chunksPerRow = (C * 4) >> 4;      // 16B chunks per row (C % 4 == 0)
    const int numTiles   = (Nrows + TILE_ROWS - 1) / TILE_ROWS;
    const int tileFloats = TILE_ROWS * C;

    // Kick off DMA for this block's first tile before touching anything else.
    int tile = blockIdx.x;
    int buf  = 0;
    if (tile < numTiles) {
        const int row0   = tile * TILE_ROWS;
        const int rows   = min(TILE_ROWS, Nrows - row0);
        const int chunks = rows * chunksPerRow;
        const char* src  = (const char*)(logits + (size_t)row0 * C);
        const unsigned dst = (unsigned)(size_t)tiles;   // low 32 bits of flat shared ptr == LDS addr
        for (int q = tid; q < chunks; q += BLOCK)
            async_copy_b128(dst + (unsigned)q * 16u, src + (size_t)q * 16);
    }

    // Zero block-local histogram while the DMA streams.
    for (int i = tid; i < 2 * MAX_CELLS; i += BLOCK) hist[i] = 0.0f;

    for (; tile < numTiles; tile += gridDim.x) {
        wait_async0();          // this wave's outstanding copies done
        __syncthreads();        // all waves' copies done -> current buf fully populated

        // Prefetch next tile into the other buffer (overlaps with compute below).
        const int nextTile = tile + gridDim.x;
        if (nextTile < numTiles) {
            const int row0   = nextTile * TILE_ROWS;
            const int rows   = min(TILE_ROWS, Nrows - row0);
            const int chunks = rows * chunksPerRow;
            const char* src  = (const char*)(logits + (size_t)row0 * C);
            const unsigned dst = (unsigned)(size_t)(tiles + (buf ^ 1) * tileFloats);
            for (int q = tid; q < chunks; q += BLOCK)
                async_copy_b128(dst + (unsigned)q * 16u, src + (size_t)q * 16);
        }

        // Process current tile from LDS: one wave32 per row.
        const int row0 = tile * TILE_ROWS;
        const int rows = min(TILE_ROWS, Nrows - row0);
        const float* tbase = tiles + buf * tileFloats;

        for (int k = wid; k < rows; k += WAVES) {
            const float* row = tbase + k * C;
            const int c0 = lane * 4;
            float x0 = -__builtin_inff(), x1 = x0, x2 = x0, x3 = x0;
            if (c0 + 3 < C) {
                float4 v = *(const float4*)(row + c0);
                x0 = v.x; x1 = v.y; x2 = v.z; x3 = v.w;
            } else {
                if (c0     < C) x0 = row[c0];
                if (c0 + 1 < C) x1 = row[c0 + 1];
                if (c0 + 2 < C) x2 = row[c0 + 2];
                if (c0 + 3 < C) x3 = row[c0 + 3];
            }

            // Per-lane max/argmax (first-index tie-break), then wave32 butterfly.
            float m = x0; int mi = c0;
            if (x1 > m) { m = x1; mi = c0 + 1; }
            if (x2 > m) { m = x2; mi = c0 + 2; }
            if (x3 > m) { m = x3; mi = c0 + 3; }
            #pragma unroll
            for (int off = 16; off > 0; off >>= 1) {
                float om = __shfl_xor(m, off, 32);
                int   oi = __shfl_xor(mi, off, 32);
                if (om > m || (om == m && oi < mi)) { m = om; mi = oi; }
            }

            // confidence = softmax max = 1 / sum(exp(x - max))
            float s = __expf(x0 - m) + __expf(x1 - m) + __expf(x2 - m) + __expf(x3 - m);
            #pragma unroll
            for (int off = 16; off > 0; off >>= 1) s += __shfl_xor(s, off, 32);

            if (lane == 0) {
                const float conf = 1.0f / s;
                const int   r    = row0 + k;
                const float acc  = (mi == labels[r]) ? 1.0f : 0.0f;
                int bin = (int)ceilf(conf * (float)nb) - 1;
                bin = bin < 0 ? 0 : (bin > nb - 1 ? nb - 1 : bin);
                const int cell = mi * nb + bin;
                if (cell < cells) {
                    atomicAdd(&hist[cell], conf);
                    atomicAdd(&hist[MAX_CELLS + cell], acc);
                }
            }
        }

        __syncthreads();   // everyone done reading buf before it is re-targeted by DMA
        buf ^= 1;
    }

    __syncthreads();
    // Flush block-local histogram into the global accumulator (skip zero cells).
    for (int i = tid; i < cells; i += BLOCK) {
        const float a = hist[i];
        const float b = hist[MAX_CELLS + i];
        if (a != 0.0f) atomicAdd(&gacc[i], a);
        if (b != 0.0f) atomicAdd(&gacc[MAX_CELLS + i], b);
    }
}

extern "C" __global__ void __launch_bounds__(BLOCK)
ece_finalize(const float* __restrict__ gacc, const int* __restrict__ nbins_p,
             int C, int Nrows, float* __restrict__ out)
{
    const int nb = *nbins_p;
    int cells = C * nb;
    if (cells > MAX_CELLS) cells = MAX_CELLS;

    float local = 0.0f;
    for (int i = threadIdx.x; i < cells; i += BLOCK)
        local += fabsf(gacc[i] - gacc[MAX_CELLS + i]);

    #pragma unroll
    for (int off = 16; off > 0; off >>= 1) local += __shfl_xor(local, off, 32);

    __shared__ float wsum[WAVES];
    if ((threadIdx.x & 31) == 0) wsum[threadIdx.x >> 5] = local;
    __syncthreads();
    if (threadIdx.x == 0) {
        float t = 0.0f;
        #pragma unroll
        for (int w = 0; w < WAVES; w++) t += wsum[w];
        out[0] = t / (float)Nrows;
    }
}

extern "C" void kernel_launch(void* const* d_in, const int* in_sizes, int n_in,
                              void* d_out, int out_size, void* d_ws, size_t ws_size,
                              hipStream_t stream) {
    const float* logits = (const float*)d_in[0];
    const int*   labels = (const int*)d_in[1];
    const int*   nbp    = (const int*)d_in[2];

    const int Nrows = in_sizes[1];                    // labels count
    const int C     = (Nrows > 0) ? (in_sizes[0] / Nrows) : 0;

    float* gacc = (float*)d_ws;
    size_t zbytes = (size_t)2 * MAX_CELLS * sizeof(float);
    if (zbytes > ws_size) zbytes = ws_size;
    hipMemsetAsync(d_ws, 0, zbytes, stream);          // capture-safe accumulator init

    const int numTiles = (Nrows + TILE_ROWS - 1) / TILE_ROWS;
    int grid = numTiles < 1024 ? numTiles : 1024;     // persistent blocks, grid-stride over tiles
    if (grid < 1) grid = 1;
    const size_t dynLds = (size_t)2 * TILE_ROWS * C * sizeof(float);

    ece_main<<<grid, BLOCK, dynLds, stream>>>(logits, labels, nbp, C, Nrows, gacc);
    ece_finalize<<<1, BLOCK, 0, stream>>>(gacc, nbp, C, Nrows, (float*)d_out);
}